// DistributedExpertAllGather_31885837205580
// MI455X (gfx1250) — compile-verified
//
#include <hip/hip_runtime.h>
#include <stdint.h>

// Problem constants (from reference)
#define E_ 8
#define C_ 4096   // GEMM M
#define K_ 1024   // GEMM K
#define N_ 4096   // GEMM N

// Tiling
#define BM 128
#define BN 128
#define BK 32
#define AST 36    // padded row stride (floats) for As[128][32]
#define BST 36    // padded row stride (floats) for Bs[128][32] (transposed: [n][k])

typedef __attribute__((ext_vector_type(2))) float v2f;
typedef __attribute__((ext_vector_type(8))) float v8f;

// ---- CDNA5 async global->LDS copies (ASYNCcnt), GVS addressing:
//      mem = SADDR(64b SGPR) + VADDR(32b per-lane byte offset); LDS = VDST(32b per-lane addr)
__device__ __forceinline__ void async_copy_b128(uint32_t lds, uint32_t voff, const void* sbase) {
  asm volatile("global_load_async_to_lds_b128 %0, %1, %2"
               :: "v"(lds), "v"(voff), "s"(sbase));
}
__device__ __forceinline__ void async_copy_b32(uint32_t lds, uint32_t voff, const void* sbase) {
  asm volatile("global_load_async_to_lds_b32 %0, %1, %2"
               :: "v"(lds), "v"(voff), "s"(sbase));
}
__device__ __forceinline__ void wait_async0() {
  asm volatile("s_wait_asynccnt 0x0" ::: "memory");
}
__device__ __forceinline__ uint32_t lds_addr(const void* p) {
  // addrspace(3) generic address: low 32 bits are the LDS byte address
  return (uint32_t)(uintptr_t)p;
}

__global__ __launch_bounds__(256) void expert_gemm_f32_wmma(
    const float* __restrict__ x,     // [E, C, K]
    const float* __restrict__ w,     // [E, K, N]
    const float* __restrict__ bias,  // [E, 1, N]
    float* __restrict__ y)           // [E, C, N]
{
  // Double-buffered LDS tiles: A is [m][k], B is stored TRANSPOSED as [n][k]
  // so a WMMA B-fragment (two consecutive K values at one N) is one b64 read.
  __shared__ float As[2][BM * AST];
  __shared__ float Bs[2][BN * BST];

  const int e  = blockIdx.z;
  const int m0 = blockIdx.y * BM;
  const int n0 = blockIdx.x * BN;

  const int tid  = threadIdx.x;
  const int wave = tid >> 5;   // wave32: 8 waves / block
  const int lane = tid & 31;
  const int half = lane >> 4;  // 0: lanes 0-15, 1: lanes 16-31
  const int l16  = lane & 15;

  const int waveM = wave & 3;   // 4 waves along M -> 32-row slabs
  const int waveN = wave >> 2;  // 2 waves along N -> 64-col slabs

  const float* Ag = x + (size_t)e * C_ * K_ + (size_t)m0 * K_;
  const float* Bg = w + (size_t)e * K_ * N_ + n0;

  v8f acc[2][4];
  #pragma unroll
  for (int f = 0; f < 2; ++f)
    #pragma unroll
    for (int g = 0; g < 4; ++g)
      acc[f][g] = (v8f){0.f, 0.f, 0.f, 0.f, 0.f, 0.f, 0.f, 0.f};

  // Issue all global->LDS copies for one K-tile (no VGPR staging, no ds stores).
  auto load_async = [&](int kt, int buf) {
    const void* sA = (const void*)(Ag + kt);                 // uniform SGPR base
    const void* sB = (const void*)(Bg + (size_t)kt * N_);    // uniform SGPR base
    #pragma unroll
    for (int i = 0; i < 4; ++i) {
      const int t = tid + i * 256;          // 0..1023
      // A tile: 128 rows x 8 float4 groups along K -> direct b128 copies
      const int r  = t >> 3;                // 0..127
      const int cg = t & 7;                 // 0..7
      async_copy_b128(lds_addr(&As[buf][r * AST + cg * 4]),
                      (uint32_t)((r * K_ + cg * 4) * 4), sA);
      // B tile: transpose folded into the copy; lane owns (n, kg) and scatters
      // 4 K-values (strided in global, contiguous-in-k in transposed LDS).
      const int n  = t & 127;               // 0..127
      const int kg = t >> 7;                // 0..7
      #pragma unroll
      for (int j = 0; j < 4; ++j)
        async_copy_b32(lds_addr(&Bs[buf][n * BST + kg * 4 + j]),
                       (uint32_t)((((kg * 4 + j) * N_) + n) * 4), sB);
    }
  };

  load_async(0, 0);
  wait_async0();
  __syncthreads();

  const int nT = K_ / BK;  // 32
  for (int t = 0; t < nT; ++t) {
    const int cur = t & 1;
    if (t + 1 < nT) load_async((t + 1) * BK, cur ^ 1);  // async DMA overlaps WMMA

    #pragma unroll
    for (int kk = 0; kk < BK; kk += 4) {
      const int krow = kk + 2 * half;
      // A fragments: 16x4 f32 (ISA 7.12.2): lane l<16 -> (M=l, K=kk..kk+1),
      // lanes 16-31 -> (M=l16, K=kk+2..kk+3). One b64 per fragment.
      v2f af[2];
      #pragma unroll
      for (int f = 0; f < 2; ++f) {
        const int row = waveM * 32 + f * 16 + l16;
        af[f] = *(const v2f*)(&As[cur][row * AST + krow]);
      }
      // B fragments: 4x16 f32, rows striped across lanes within a VGPR:
      // contiguous b64 per fragment thanks to the transposed LDS layout.
      v2f bf[4];
      #pragma unroll
      for (int g = 0; g < 4; ++g) {
        const int col = waveN * 64 + g * 16 + l16;
        bf[g] = *(const v2f*)(&Bs[cur][col * BST + krow]);
      }
      #pragma unroll
      for (int f = 0; f < 2; ++f)
        #pragma unroll
        for (int g = 0; g < 4; ++g)
          acc[f][g] = __builtin_amdgcn_wmma_f32_16x16x4_f32(
              /*neg_a=*/false, af[f], /*neg_b=*/false, bf[g],
              /*c_mod=*/(short)0, acc[f][g],
              /*reuse_a=*/false, /*reuse_b=*/false);
    }

    if (t + 1 < nT) {
      wait_async0();     // this wave's tile(t+1) copies are done
      __syncthreads();   // all waves' copies visible; readers flip buffers
    }
  }

  // Epilogue: C/D layout -> element (VGPR r, lane) = (M = r + 8*half, N = l16)
  const float* be = bias + (size_t)e * N_;
  float*       Yg = y + (size_t)e * C_ * N_;
  #pragma unroll
  for (int g = 0; g < 4; ++g) {
    const int gn = n0 + waveN * 64 + g * 16 + l16;
    const float bv = be[gn];
    #pragma unroll
    for (int f = 0; f < 2; ++f) {
      const int mbase = m0 + waveM * 32 + f * 16 + 8 * half;
      #pragma unroll
      for (int r = 0; r < 8; ++r) {
        Yg[(size_t)(mbase + r) * N_ + gn] = acc[f][g][r] + bv;
      }
    }
  }
}

extern "C" void kernel_launch(void* const* d_in, const int* in_sizes, int n_in,
                              void* d_out, int out_size, void* d_ws, size_t ws_size,
                              hipStream_t stream) {
  const float* x = (const float*)d_in[0];  // [E, C, M_IN] fp32
  const float* w = (const float*)d_in[1];  // [E, M_IN, N_OUT] fp32
  const float* b = (const float*)d_in[2];  // [E, 1, N_OUT] fp32
  float* y = (float*)d_out;                // [E, C, N_OUT] fp32

  dim3 grid(N_ / BN, C_ / BM, E_);  // 32 x 32 x 8
  expert_gemm_f32_wmma<<<grid, 256, 0, stream>>>(x, w, b, y);
}